// WorldModel_3642132267800
// MI455X (gfx1250) — compile-verified
//
#include <hip/hip_runtime.h>
#include <hip/hip_bf16.h>
#include <math.h>

typedef __attribute__((ext_vector_type(2))) float v2f;
typedef __attribute__((ext_vector_type(8))) float v8f;

#define WMMA4(Af, Bf, Cf) \
  __builtin_amdgcn_wmma_f32_16x16x4_f32(false, (Af), false, (Bf), (short)0, (Cf), false, false)

// ---------------- constants (reference shapes) ----------------
#define BATCH 1024
#define TSEQ  129      // T
#define OBS   14
#define HID   128
#define DDIM  200
#define SDIM  30
#define ADIM  6
#define OUTW  135      // 14 + 1 + 4*30

__device__ __forceinline__ float softplusf(float x) {
  return x > 20.f ? x : log1pf(expf(x));
}

__device__ __forceinline__ unsigned pcg_hash(unsigned x) {
  x = x * 747796405u + 2891336453u;
  unsigned w = ((x >> ((x >> 28u) + 4u)) ^ x) * 277803737u;
  return (w >> 22u) ^ w;
}

__device__ __forceinline__ float gauss_eps(int seed, int t, int b, int s) {
  unsigned base = (unsigned)seed * 2654435761u ^ ((unsigned)t * 9781u)
                ^ ((unsigned)b * 31337u) ^ ((unsigned)s * 6271u);
  unsigned h1 = pcg_hash(base);
  unsigned h2 = pcg_hash(h1 + 0x9E3779B9u);
  float u1 = ((float)h1 + 0.5f) * (1.0f / 4294967296.0f);
  float u2 = ((float)h2 + 0.5f) * (1.0f / 4294967296.0f);
  return sqrtf(-2.f * logf(u1)) * cosf(6.28318530717958647f * u2);
}

// ================= encoder: embed = relu(obs@w1+b1)@w2+b2 ================
// 128 threads = 4 waves, one 16-token M-tile per wave.  LDS = 40 KB.
__global__ void wm_encoder(const float* __restrict__ obs,
                           const float* __restrict__ w1, const float* __restrict__ b1,
                           const float* __restrict__ w2, const float* __restrict__ b2,
                           float* __restrict__ embed) {
  __shared__ v2f   sW1[8][HID];        // paired rows of w1 (K=14 padded to 16), 8 KB
  __shared__ float hid[4][16][HID];    // per-wave hidden tiles, 32 KB
  const int tid  = threadIdx.x;
  const int wave = tid >> 5;
  const int lane = tid & 31;
  const int half = lane >> 4;
  const int lm   = lane & 15;
  const int m0   = (blockIdx.x * 4 + wave) * 16;
  const int arow = m0 + lm;

  // stage paired w1: entry (k2,c) = {w1[2k2][c], w1[2k2+1][c]} (rows>=14 -> 0)
  for (int idx = tid; idx < 8 * HID; idx += 128) {
    const int k2 = idx >> 7, c = idx & (HID - 1);
    const int kk = 2 * k2;
    v2f p;
    p.x = (kk     < OBS) ? w1[kk * HID + c]       : 0.f;
    p.y = (kk + 1 < OBS) ? w1[(kk + 1) * HID + c] : 0.f;
    sW1[k2][c] = p;
  }
  __syncthreads();

  const float* orow = obs + (size_t)arow * OBS;

  // layer 1: K=14 padded to 16
  for (int nt = 0; nt < HID / 16; ++nt) {
    const int col = nt * 16 + lm;
    v8f acc = {};
#pragma unroll
    for (int kt = 0; kt < 4; ++kt) {
      const int k  = kt * 4 + 2 * half;
      const int kc = k < 12 ? k : 12;              // clamped, always valid
      v2f a = *(const v2f*)(orow + kc);            // global_load_b64
      if (k >= OBS) { a.x = 0.f; a.y = 0.f; }      // k==14 pad (branchless)
      const v2f b = sW1[(kt * 2 + half)][col];     // ds_load_b64
      acc = WMMA4(a, b, acc);
    }
    const float bias = b1[col];
#pragma unroll
    for (int v = 0; v < 8; ++v) {
      const float x = acc[v] + bias;
      hid[wave][v + 8 * half][col] = x > 0.f ? x : 0.f;
    }
  }
  __syncthreads();

  // layer 2: K=128 exact
  for (int nt = 0; nt < HID / 16; ++nt) {
    const int col = nt * 16 + lm;
    v8f acc = {};
#pragma unroll 2
    for (int kt = 0; kt < HID / 4; ++kt) {
      const int k = kt * 4 + 2 * half;
      const v2f a = *(const v2f*)&hid[wave][lm][k];  // ds_load_b64
      v2f b;
      b.x = w2[(k    ) * HID + col];
      b.y = w2[(k + 1) * HID + col];
      acc = WMMA4(a, b, acc);
    }
    const float bias = b2[col];
#pragma unroll
    for (int v = 0; v < 8; ++v)
      embed[(size_t)(m0 + v + 8 * half) * HID + col] = acc[v] + bias;
  }
}

// ================= GRU step =========================================
// grid (13, 8): blockIdx.x = N-tile of D, 8 waves each take an M-tile.
// All waves share one N-tile -> gate weights staged paired in LDS (45.3 KB).
__global__ void wm_gru_step(const float* __restrict__ zprev,   // (B,30)
                            const float* __restrict__ hprev,   // (B,200)
                            const int*   __restrict__ action,  // (B,T)
                            int t,
                            const float* __restrict__ wi, const float* __restrict__ bi,
                            const float* __restrict__ wh, const float* __restrict__ bn,
                            float* __restrict__ hnew) {        // (B,200)
  __shared__ v2f sWi[3][18][16];    // paired rows of wi per gate (K=36)
  __shared__ v2f sWh[3][100][16];   // paired rows of wh per gate (K=200)
  const int tid  = threadIdx.x;
  const int wave = tid >> 5;
  const int lane = tid & 31;
  const int half = lane >> 4;
  const int lm   = lane & 15;
  const int n0   = blockIdx.x * 16;
  const int m0   = (blockIdx.y * 8 + wave) * 16;
  const int arow = m0 + lm;
  const int col  = n0 + lm;
  const bool cok = col < DDIM;

  // ---- stage paired weights (clamped col, zero-select) ----
  for (int g = 0; g < 3; ++g) {
    const int off = g * DDIM;
    for (int idx = tid; idx < 18 * 16; idx += 256) {
      const int k2 = idx >> 4, c = idx & 15;
      const int cc  = n0 + c;
      const int ccc = cc < DDIM ? cc : DDIM - 1;
      const int kk  = 2 * k2;
      v2f p;
      const float px = wi[(kk < 35 ? kk : 35) * 3 * DDIM + off + ccc];
      const float py = wi[(kk + 1 < 36 ? kk + 1 : 35) * 3 * DDIM + off + ccc];
      p.x = (cc < DDIM && kk     < 36) ? px : 0.f;
      p.y = (cc < DDIM && kk + 1 < 36) ? py : 0.f;
      sWi[g][k2][c] = p;
    }
    for (int idx = tid; idx < 100 * 16; idx += 256) {
      const int k2 = idx >> 4, c = idx & 15;
      const int cc  = n0 + c;
      const int ccc = cc < DDIM ? cc : DDIM - 1;
      const int kk  = 2 * k2;
      v2f p;
      p.x = (cc < DDIM) ? wh[kk * 3 * DDIM + off + ccc]       : 0.f;
      p.y = (cc < DDIM) ? wh[(kk + 1) * 3 * DDIM + off + ccc] : 0.f;
      sWh[g][k2][c] = p;
    }
  }
  __syncthreads();

  const int act = action[(size_t)arow * TSEQ + t];
  const float* zrowA = zprev + (size_t)arow * SDIM;
  const float* hrowA = hprev + (size_t)arow * DDIM;

  v8f accR = {}, accZ = {}, accNi = {}, accNh = {};

  // gi = x @ wi, x = [z_prev | onehot(a)], K = 36
#pragma unroll
  for (int kt = 0; kt < 9; ++kt) {
    const int k  = kt * 4 + 2 * half;
    const int k2 = kt * 2 + half;
    const v2f zp = *(const v2f*)(zrowA + (k < 28 ? k : 28));   // always valid
    v2f a;
    a.x = (k     < SDIM) ? zp.x : ((k     - SDIM) == act ? 1.f : 0.f);
    a.y = (k + 1 < SDIM) ? zp.y : ((k + 1 - SDIM) == act ? 1.f : 0.f);
    accR  = WMMA4(a, sWi[0][k2][lm], accR);
    accZ  = WMMA4(a, sWi[1][k2][lm], accZ);
    accNi = WMMA4(a, sWi[2][k2][lm], accNi);
  }
  // gh = h @ wh, K = 200
#pragma unroll 2
  for (int kt = 0; kt < 50; ++kt) {
    const int k  = kt * 4 + 2 * half;
    const int k2 = kt * 2 + half;
    const v2f a = *(const v2f*)(hrowA + k);                    // global_load_b64
    accR  = WMMA4(a, sWh[0][k2][lm], accR);
    accZ  = WMMA4(a, sWh[1][k2][lm], accZ);
    accNh = WMMA4(a, sWh[2][k2][lm], accNh);
  }

  if (cok) {
    const float biR = bi[col], biZ = bi[DDIM + col], biN = bi[2 * DDIM + col];
    const float bnc = bn[col];
#pragma unroll
    for (int v = 0; v < 8; ++v) {
      const int row = m0 + v + 8 * half;
      const float r = 1.f / (1.f + expf(-(accR[v] + biR)));
      const float u = 1.f / (1.f + expf(-(accZ[v] + biZ)));
      const float n = tanhf(accNi[v] + biN + r * (accNh[v] + bnc));
      const float hp = hprev[(size_t)row * DDIM + col];
      hnew[(size_t)row * DDIM + col] = (1.f - u) * n + u * hp;
    }
  }
}

// ================= heads: pm/ps/qm/qs + z sample ==========================
// grid (2, 8): blockIdx.x = N-tile of S, 8 waves each take an M-tile.
// qm/qs (K=328) staged paired in LDS (42 KB); pm/ps branchless from global.
__global__ void wm_heads(const float* __restrict__ hnew,   // (B,200)
                         const float* __restrict__ embed,  // (B,T,128)
                         int t,
                         const float* __restrict__ pmw, const float* __restrict__ pmb,
                         const float* __restrict__ psw, const float* __restrict__ psb,
                         const float* __restrict__ qmw, const float* __restrict__ qmb,
                         const float* __restrict__ qsw, const float* __restrict__ qsb,
                         const int* __restrict__ ols_p, const int* __restrict__ seed_p,
                         float* __restrict__ znew,         // (B,30)
                         float* __restrict__ out) {        // (B,T-1,135)
  __shared__ v2f sQ[2][164][16];    // paired rows of qm_w / qs_w (K=328)
  const int tid  = threadIdx.x;
  const int wave = tid >> 5;
  const int lane = tid & 31;
  const int half = lane >> 4;
  const int lm   = lane & 15;
  const int n0   = blockIdx.x * 16;
  const int m0   = (blockIdx.y * 8 + wave) * 16;
  const int arow = m0 + lm;
  const int col  = n0 + lm;
  const bool cok = col < SDIM;
  const int colc = col < SDIM ? col : SDIM - 1;

  // ---- stage paired qm/qs ----
  for (int m = 0; m < 2; ++m) {
    const float* qw = m ? qsw : qmw;
    for (int idx = tid; idx < 164 * 16; idx += 256) {
      const int k2 = idx >> 4, c = idx & 15;
      const int cc  = n0 + c;
      const int ccc = cc < SDIM ? cc : SDIM - 1;
      const int kk  = 2 * k2;
      v2f p;
      p.x = (cc < SDIM) ? qw[kk * SDIM + ccc]       : 0.f;
      p.y = (cc < SDIM) ? qw[(kk + 1) * SDIM + ccc] : 0.f;
      sQ[m][k2][c] = p;
    }
  }
  __syncthreads();

  const float* hrowA = hnew + (size_t)arow * DDIM;
  const float* erowA = embed + ((size_t)arow * TSEQ + (t + 1)) * HID;

  v8f aPM = {}, aPS = {}, aQM = {}, aQS = {};

  // shared K over h_new: pm, ps, qm, qs (rows 0..199)
#pragma unroll 2
  for (int kt = 0; kt < 50; ++kt) {
    const int k  = kt * 4 + 2 * half;
    const int k2 = kt * 2 + half;
    const v2f a = *(const v2f*)(hrowA + k);
    v2f b0, b1;
    const float p0 = pmw[(k    ) * SDIM + colc];
    const float p1 = pmw[(k + 1) * SDIM + colc];
    const float s0 = psw[(k    ) * SDIM + colc];
    const float s1 = psw[(k + 1) * SDIM + colc];
    b0.x = cok ? p0 : 0.f;  b0.y = cok ? p1 : 0.f;
    b1.x = cok ? s0 : 0.f;  b1.y = cok ? s1 : 0.f;
    aPM = WMMA4(a, b0, aPM);
    aPS = WMMA4(a, b1, aPS);
    aQM = WMMA4(a, sQ[0][k2][lm], aQM);
    aQS = WMMA4(a, sQ[1][k2][lm], aQS);
  }
  // qm/qs tail over emb (rows 200..327)
#pragma unroll 2
  for (int kt = 0; kt < 32; ++kt) {
    const int k  = kt * 4 + 2 * half;
    const int k2 = 100 + kt * 2 + half;
    const v2f a = *(const v2f*)(erowA + k);
    aQM = WMMA4(a, sQ[0][k2][lm], aQM);
    aQS = WMMA4(a, sQ[1][k2][lm], aQS);
  }

  if (cok) {
    const int ols  = *ols_p;
    const int seed = *seed_p;
    const float bpm = pmb[col], bps = psb[col], bqm = qmb[col], bqs = qsb[col];
#pragma unroll
    for (int v = 0; v < 8; ++v) {
      const int row = m0 + v + 8 * half;               // batch index
      const float pm = aPM[v] + bpm;
      const float ps = softplusf(aPS[v] + bps) + 0.001f;
      const float qm = aQM[v] + bqm;
      const float qs = softplusf(aQS[v] + bqs) + 0.001f;
      const float e  = gauss_eps(seed, t, row, col);
      const float z  = (t < ols) ? (qm + qs * e) : (pm + ps * e);
      znew[(size_t)row * SDIM + col] = z;
      float* o = out + ((size_t)row * (TSEQ - 1) + t) * OUTW;
      o[15 + col]  = pm;
      o[45 + col]  = ps;
      o[75 + col]  = qm;
      o[105 + col] = qs;
    }
  }
}

// ================= decoder + reward, fused 256-wide hidden ===============
// 64 threads = 2 waves per block; LDS = 48 KB (hid 32 KB + paired W2cat 16 KB)
__global__ void wm_decoder(const float* __restrict__ hws,  // (T,B,200) slots
                           const float* __restrict__ zws,  // (T,B,30)  slots
                           const float* __restrict__ dw1, const float* __restrict__ db1,
                           const float* __restrict__ dw2, const float* __restrict__ db2,
                           const float* __restrict__ rw1, const float* __restrict__ rb1,
                           const float* __restrict__ rw2, const float* __restrict__ rb2,
                           float* __restrict__ out) {
  __shared__ float hid[2][16][256];   // 32 KB
  __shared__ v2f   sW2[128][16];      // paired block-diag [dec_w2 ; rew_w2], 16 KB
  const int tid  = threadIdx.x;
  const int wave = tid >> 5;
  const int lane = tid & 31;
  const int half = lane >> 4;
  const int lm   = lane & 15;
  const int m0   = (blockIdx.x * 2 + wave) * 16;
  const int Tm1  = TSEQ - 1;

  // stage paired block-diagonal second-layer weights (K=256, N=16)
  for (int idx = tid; idx < 128 * 16; idx += 64) {
    const int k2 = idx >> 4, n = idx & 15;
    const int kk = 2 * k2;
    float e0 = 0.f, e1 = 0.f;
    if (n < OBS) {
      if (kk < HID) { e0 = dw2[kk * OBS + n]; e1 = dw2[(kk + 1) * OBS + n]; }
    } else if (n == OBS) {
      if (kk >= HID) { e0 = rw2[kk - HID]; e1 = rw2[kk + 1 - HID]; }
    }
    v2f p; p.x = e0; p.y = e1;
    sW2[k2][n] = p;
  }
  __syncthreads();

  const int tok = m0 + lm;
  const int bb  = tok / Tm1;
  const int tt  = tok - bb * Tm1;
  const float* hrow = hws + ((size_t)(tt + 1) * BATCH + bb) * DDIM;
  const float* zrow = zws + ((size_t)(tt + 1) * BATCH + bb) * SDIM;

  // phase 1: hidden(256) = relu(hz @ [dec_w1 | rew_w1] + [db1 | rb1]); K=230
  for (int nt = 0; nt < 16; ++nt) {
    const int hc = nt * 16 + lm;                 // 0..255
    const float* W  = (hc < HID) ? dw1 : rw1;
    const int    c  = hc & (HID - 1);
    v8f acc = {};
#pragma unroll 2
    for (int kt = 0; kt < 58; ++kt) {
      const int k = kt * 4 + 2 * half;
      // A: hz pair via pointer select (always-valid clamped address)
      const int kz = k - DDIM;
      const float* p = (k < DDIM) ? (hrow + k) : (zrow + (kz < SDIM ? kz : SDIM - 2));
      v2f a = *(const v2f*)p;
      if (k >= DDIM + SDIM) { a.x = 0.f; a.y = 0.f; }   // k==230 pad
      // B: branchless clamped loads
      const int kc0 = k     < DDIM + SDIM ? k     : DDIM + SDIM - 1;
      const int kc1 = k + 1 < DDIM + SDIM ? k + 1 : DDIM + SDIM - 1;
      v2f b;
      const float bx = W[kc0 * HID + c];
      const float by = W[kc1 * HID + c];
      b.x = (k     < DDIM + SDIM) ? bx : 0.f;
      b.y = (k + 1 < DDIM + SDIM) ? by : 0.f;
      acc = WMMA4(a, b, acc);
    }
    const float bias = (hc < HID) ? db1[c] : rb1[c];
#pragma unroll
    for (int v = 0; v < 8; ++v) {
      const float x = acc[v] + bias;
      hid[wave][v + 8 * half][hc] = x > 0.f ? x : 0.f;
    }
  }
  __syncthreads();

  // phase 2: [recon(14) | rew(1)] = hidden @ sW2; K=256 exact
  {
    v8f acc = {};
#pragma unroll 2
    for (int kt = 0; kt < 64; ++kt) {
      const int k  = kt * 4 + 2 * half;
      const int k2 = kt * 2 + half;
      const v2f a = *(const v2f*)&hid[wave][lm][k];
      acc = WMMA4(a, sW2[k2][lm], acc);
    }
#pragma unroll
    for (int v = 0; v < 8; ++v) {
      const int tok2 = m0 + v + 8 * half;
      const int b2 = tok2 / Tm1;
      const int t2 = tok2 - b2 * Tm1;
      float* o = out + ((size_t)b2 * Tm1 + t2) * OUTW;
      if (lm < OBS)       o[lm]  = acc[v] + db2[lm];
      else if (lm == OBS) o[OBS] = acc[v] + rb2[0];
    }
  }
}

// ============================ launcher ===================================
extern "C" void kernel_launch(void* const* d_in, const int* in_sizes, int n_in,
                              void* d_out, int out_size, void* d_ws, size_t ws_size,
                              hipStream_t stream) {
  const float* obs     = (const float*)d_in[0];
  const int*   action  = (const int*)d_in[1];
  const float* enc_w1  = (const float*)d_in[2];
  const float* enc_b1  = (const float*)d_in[3];
  const float* enc_w2  = (const float*)d_in[4];
  const float* enc_b2  = (const float*)d_in[5];
  const float* gru_wi  = (const float*)d_in[6];
  const float* gru_bi  = (const float*)d_in[7];
  const float* gru_wh  = (const float*)d_in[8];
  const float* gru_bn  = (const float*)d_in[9];
  const float* pm_w    = (const float*)d_in[10];
  const float* pm_b    = (const float*)d_in[11];
  const float* ps_w    = (const float*)d_in[12];
  const float* ps_b    = (const float*)d_in[13];
  const float* qm_w    = (const float*)d_in[14];
  const float* qm_b    = (const float*)d_in[15];
  const float* qs_w    = (const float*)d_in[16];
  const float* qs_b    = (const float*)d_in[17];
  const float* dec_w1  = (const float*)d_in[18];
  const float* dec_b1  = (const float*)d_in[19];
  const float* dec_w2  = (const float*)d_in[20];
  const float* dec_b2  = (const float*)d_in[21];
  const float* rew_w1  = (const float*)d_in[22];
  const float* rew_b1  = (const float*)d_in[23];
  const float* rew_w2  = (const float*)d_in[24];
  const float* rew_b2  = (const float*)d_in[25];
  const int*   ols     = (const int*)d_in[26];
  const int*   seed    = (const int*)d_in[27];
  float* out = (float*)d_out;

  // workspace partition (floats)
  float* embed = (float*)d_ws;                                   // B*T*HID
  float* hws   = embed + (size_t)BATCH * TSEQ * HID;             // T slots of (B,200)
  float* zws   = hws   + (size_t)TSEQ * BATCH * DDIM;            // T slots of (B,30)

  // zero h0 / z0 (slot 0)
  hipMemsetAsync(hws, 0, (size_t)BATCH * DDIM * sizeof(float), stream);
  hipMemsetAsync(zws, 0, (size_t)BATCH * SDIM * sizeof(float), stream);

  // encoder: (B*T)/16 tiles, 4 waves/block
  wm_encoder<<<(BATCH * TSEQ) / 64, 128, 0, stream>>>(
      obs, enc_w1, enc_b1, enc_w2, enc_b2, embed);

  // sequential scan
  for (int t = 0; t < TSEQ - 1; ++t) {
    const float* zprev = zws + (size_t)t * BATCH * SDIM;
    const float* hprev = hws + (size_t)t * BATCH * DDIM;
    float* hnew = hws + (size_t)(t + 1) * BATCH * DDIM;
    float* znew = zws + (size_t)(t + 1) * BATCH * SDIM;

    wm_gru_step<<<dim3(13, 8), 256, 0, stream>>>(zprev, hprev, action, t,
                                                 gru_wi, gru_bi, gru_wh, gru_bn, hnew);

    wm_heads<<<dim3(2, 8), 256, 0, stream>>>(hnew, embed, t,
                                             pm_w, pm_b, ps_w, ps_b,
                                             qm_w, qm_b, qs_w, qs_b,
                                             ols, seed, znew, out);
  }

  // decoder + reward heads: 8192 token tiles, 2 waves/block
  wm_decoder<<<4096, 64, 0, stream>>>(hws, zws,
                                      dec_w1, dec_b1, dec_w2, dec_b2,
                                      rew_w1, rew_b1, rew_w2, rew_b2, out);

  (void)in_sizes; (void)n_in; (void)out_size; (void)ws_size;
}